// GlobalFilter_80255758893130
// MI455X (gfx1250) — compile-verified
//
#include <hip/hip_runtime.h>
#include <math.h>

typedef __attribute__((ext_vector_type(2))) float v2f;
typedef __attribute__((ext_vector_type(4))) float v4f;
typedef __attribute__((ext_vector_type(8))) float v8f;

#define WMMA4(A, B, C) \
  __builtin_amdgcn_wmma_f32_16x16x4_f32(false, (A), false, (B), (short)0, (C), false, false)

// LDS pool offsets (floats). Buffers alias across disjoint lifetimes:
//  stage1: reads Xs, writes Ar/Ai      stage2: reads Ar/Ai, writes Zr/Zi (over dead Xs)
//  stage4: reads Zr/Zi, writes Tr/Ti (in Ar/Ai)   stage5: reads Tr/Ti, writes Out (over dead Z)
#define AR_OFF 0      // [u16][w16][c16] 4096 ; later Tr [h16][v8][c16] 2048
#define AI_OFF 4096   // same (imag) ; later Ti
#define XS_OFF 8192   // [h16][w14][c16] 3584 ; later Out [h14][w16][c16] 3584
#define ZR_OFF 8192   // [u16][v8][c16] 2048 (overlays dead Xs)
#define ZI_OFF 10240  // [u16][v8][c16] 2048
#define POOL_N 12288  // 48 KB

__global__ __launch_bounds__(32) void gf_kernel(const float* __restrict__ x,
                                                const float* __restrict__ wgt,
                                                float* __restrict__ out) {
  const int lane = threadIdx.x;
  const int n  = lane & 15;   // channel-in-tile (N dim across lanes)
  const int hh = lane >> 4;   // 0: rows j / K lo half, 1: rows j+8 / K hi half
  const int cbase = blockIdx.x * 16;  // channel tile
  const int b = blockIdx.y;           // batch

  __shared__ __align__(16) float P[POOL_N];
  __shared__ float ctab[16];  // cos(2*pi*r/14)/sqrt(14)
  __shared__ float stab[16];  // sin(2*pi*r/14)/sqrt(14)

  float* ArL = P + AR_OFF;
  float* AiL = P + AI_OFF;
  float* XsL = P + XS_OFF;
  float* ZrL = P + ZR_OFF;
  float* ZiL = P + ZI_OFF;

  const v4f z4 = {0.f, 0.f, 0.f, 0.f};

  // ---- trig table (lanes 0..13), A-pad zeroing, X tile load ----
  if (lane < 14) {
    float a = (float)lane * 0.44879895051282760549f; // 2*pi/14
    const float S = 0.26726124191242439f;            // 1/sqrt(14)
    ctab[lane] = cosf(a) * S;
    stab[lane] = sinf(a) * S;
  }
  // Xs rows h=14,15 zero (3136..3583, contiguous)
  for (int t = lane; t < 112; t += 32) *(v4f*)&XsL[3136 + t * 4] = z4;
  // ArL/AiL w=14,15 pads: per u, 32 floats at u*256+224
  for (int t = lane; t < 128; t += 32) {
    int u = t >> 3, r4 = (t & 7) * 4;
    *(v4f*)&ArL[u * 256 + 224 + r4] = z4;
    *(v4f*)&AiL[u * 256 + 224 + r4] = z4;
  }
  const size_t xbase = (size_t)b * 196 * 768 + cbase;
  for (int i = lane * 4; i < 3136; i += 128) {
    int pos = i >> 4, c = i & 15;
    *(v4f*)&XsL[i] =
        __builtin_nontemporal_load((const v4f*)&x[xbase + (size_t)pos * 768 + c]);
  }
  __syncthreads();

  // unconditional fragment builders (K pads are zero rows)
  auto bfrag = [&](const float* buf, int q, int stride) -> v2f {
    int k0 = 4 * q + 2 * hh;
    v2f r;
    r.x = buf[(k0)     * stride + n];
    r.y = buf[(k0 + 1) * stride + n];
    return r;
  };
  auto afrag = [&](auto&& f, int q) -> v2f {
    int k0 = 4 * q + 2 * hh;
    v2f r;
    r.x = f(n, k0);
    r.y = f(n, k0 + 1);
    return r;
  };
  auto tcos = [&](int m, int k) { return ctab[(m * k) % 14]; };
  auto tsin = [&](int m, int k) { return stab[(m * k) % 14]; };

  const v8f vz = {0.f, 0.f, 0.f, 0.f, 0.f, 0.f, 0.f, 0.f};

  // ========== Stage 1: FFT along H.  Ar = Fc X ; Ai = Fs X (Fs = -sin) ==========
  {
    v2f fc[4], fs[4];
#pragma unroll
    for (int q = 0; q < 4; ++q) {
      fc[q] = afrag([&](int m, int k) { return  tcos(m, k); }, q);
      fs[q] = afrag([&](int m, int k) { return -tsin(m, k); }, q);
    }
    for (int w = 0; w < 14; ++w) {
      v2f xb[4];
#pragma unroll
      for (int q = 0; q < 4; ++q) xb[q] = bfrag(XsL + w * 16, q, 224);
      v8f ar = vz, ai = vz;  // 2 independent WMMA chains
#pragma unroll
      for (int q = 0; q < 4; ++q) {
        ar = WMMA4(fc[q], xb[q], ar);
        ai = WMMA4(fs[q], xb[q], ai);
      }
#pragma unroll
      for (int j = 0; j < 8; ++j) {
        int u = j + 8 * hh;  // 0..15 unconditional; rows 14,15 never read
        ArL[u * 256 + w * 16 + n] = ar[j];
        AiL[u * 256 + w * 16 + n] = ai[j];
      }
    }
  }
  __syncthreads();

  // ====== Stage 2: rfft along W, stacked: [Zr;Zi] = [Gc;Gs]Ar + [-Gs;Gc]Ai ======
  {
    v2f g1[4], g2[4];
#pragma unroll
    for (int q = 0; q < 4; ++q) {
      g1[q] = afrag([&](int m, int k) {
        return m < 8 ? tcos(m, k) : -tsin(m - 8, k);   // Gc ; Gs
      }, q);
      g2[q] = afrag([&](int m, int k) {
        return m < 8 ? tsin(m, k) : tcos(m - 8, k);    // -Gs ; Gc
      }, q);
    }
    for (int u = 0; u < 14; ++u) {
      v2f br[4], bi[4];
#pragma unroll
      for (int q = 0; q < 4; ++q) {
        br[q] = bfrag(ArL + u * 256, q, 16);
        bi[q] = bfrag(AiL + u * 256, q, 16);
      }
      v8f z1 = vz, z2 = vz;  // 2 independent chains, combined at the end
#pragma unroll
      for (int q = 0; q < 4; ++q) {
        z1 = WMMA4(g1[q], br[q], z1);
        z2 = WMMA4(g2[q], bi[q], z2);
      }
      v8f z = z1 + z2;
      float* dst = hh ? ZiL : ZrL;  // rows 0-7 = Zr[v], rows 8-15 = Zi[v]
#pragma unroll
      for (int j = 0; j < 8; ++j) dst[u * 128 + j * 16 + n] = z[j];
    }
    // zero Zr/Zi u=14,15 pads (this space held X data until stage 2)
    for (int t = lane; t < 64; t += 32) {
      *(v4f*)&ZrL[1792 + t * 4] = z4;
      *(v4f*)&ZiL[1792 + t * 4] = z4;
    }
  }
  __syncthreads();

  // ===== Stage 3: pointwise complex multiply by weight (in place, vec4) =====
  for (int i = lane * 4; i < 1792; i += 128) {
    int uv = i >> 4, c = i & 15;
    v4f zr = *(const v4f*)&ZrL[i];
    v4f zi = *(const v4f*)&ZiL[i];
    size_t wb = ((size_t)uv * 768 + cbase + c) * 2;
    v4f wa = *(const v4f*)&wgt[wb];      // wr0,wi0,wr1,wi1
    v4f wc = *(const v4f*)&wgt[wb + 4];  // wr2,wi2,wr3,wi3
    v4f wr = {wa.x, wa.z, wc.x, wc.z};
    v4f wi = {wa.y, wa.w, wc.y, wc.w};
    v4f yr = zr * wr - zi * wi;
    v4f yi = zr * wi + zi * wr;
    *(v4f*)&ZrL[i] = yr;
    *(v4f*)&ZiL[i] = yi;
  }
  __syncthreads();

  // ======= Stage 4: ifft along H.  Tr = Bc Yr - Bs Yi ; Ti = Bc Yi + Bs Yr =======
  float* TrL = ArL;  // [h16][v8][c16]
  float* TiL = AiL;
  {
    v2f bc[4], bsp[4], bsn[4];
#pragma unroll
    for (int q = 0; q < 4; ++q) {
      bc[q]  = afrag([&](int m, int k) { return  tcos(m, k); }, q);
      bsp[q] = afrag([&](int m, int k) { return  tsin(m, k); }, q);
      bsn[q] = -bsp[q];
    }
    for (int v = 0; v < 8; ++v) {
      v2f yr[4], yi[4];
#pragma unroll
      for (int q = 0; q < 4; ++q) {
        yr[q] = bfrag(ZrL + v * 16, q, 128);
        yi[q] = bfrag(ZiL + v * 16, q, 128);
      }
      v8f t1 = vz, t2 = vz, t3 = vz, t4 = vz;  // 4 independent chains
#pragma unroll
      for (int q = 0; q < 4; ++q) {
        t1 = WMMA4(bc[q],  yr[q], t1);
        t2 = WMMA4(bsn[q], yi[q], t2);
        t3 = WMMA4(bc[q],  yi[q], t3);
        t4 = WMMA4(bsp[q], yr[q], t4);
      }
      v8f tr = t1 + t2, ti = t3 + t4;
#pragma unroll
      for (int j = 0; j < 8; ++j) {
        int h = j + 8 * hh;  // 0..15 unconditional; rows 14,15 never read
        TrL[h * 128 + v * 16 + n] = tr[j];
        TiL[h * 128 + v * 16 + n] = ti[j];
      }
    }
  }
  __syncthreads();

  // ===== Stage 5: c2r along W. out = Cr Tr + Ci Ti (Ci cols v=0,7 are zero) =====
  float* OutL = XsL;  // [h14][w16][c16] (Zr/Zi dead now)
  {
    v2f cr[2], ci[2];
#pragma unroll
    for (int q = 0; q < 2; ++q) {
      cr[q] = afrag([&](int m, int k) {
        float t = tcos(m, k);
        return (k == 0 || k == 7) ? t : 2.0f * t;
      }, q);
      ci[q] = afrag([&](int m, int k) {
        float t = -tsin(m, k);
        return (k == 0 || k == 7) ? t : 2.0f * t;
      }, q);
    }
    for (int h = 0; h < 14; ++h) {
      v2f tr[2], ti[2];
#pragma unroll
      for (int q = 0; q < 2; ++q) {
        tr[q] = bfrag(TrL + h * 128, q, 16);
        ti[q] = bfrag(TiL + h * 128, q, 16);
      }
      v8f o1 = vz, o2 = vz;  // 2 independent chains
#pragma unroll
      for (int q = 0; q < 2; ++q) {
        o1 = WMMA4(cr[q], tr[q], o1);
        o2 = WMMA4(ci[q], ti[q], o2);
      }
      v8f o = o1 + o2;
#pragma unroll
      for (int j = 0; j < 8; ++j) {
        int w = j + 8 * hh;  // 0..15 unconditional; w=14,15 staged but not copied
        OutL[h * 256 + w * 16 + n] = o[j];
      }
    }
  }
  __syncthreads();

  // ---- coalesced vec4 copy LDS -> global, non-temporal (drops w=14,15 pad) ----
  for (int i = lane * 4; i < 3136; i += 128) {
    int h = i / 224;
    int rem = i - h * 224;          // w*16 + c
    int pos = h * 14 + (rem >> 4);  // h*14 + w
    v4f v = *(const v4f*)&OutL[h * 256 + rem];
    __builtin_nontemporal_store(v, (v4f*)&out[xbase + (size_t)pos * 768 + (rem & 15)]);
  }
}

extern "C" void kernel_launch(void* const* d_in, const int* in_sizes, int n_in,
                              void* d_out, int out_size, void* d_ws, size_t ws_size,
                              hipStream_t stream) {
  (void)in_sizes; (void)n_in; (void)d_ws; (void)ws_size; (void)out_size;
  const float* x   = (const float*)d_in[0];  // [256,196,768] f32
  const float* wgt = (const float*)d_in[1];  // [14,8,768,2] f32
  float* out = (float*)d_out;                // [256,196,768] f32
  dim3 grid(48, 256);  // (C/16 tiles, B)
  gf_kernel<<<grid, dim3(32, 1, 1), 0, stream>>>(x, wgt, out);
}